// MPNN_35012573397770
// MI455X (gfx1250) — compile-verified
//
#include <hip/hip_runtime.h>

typedef __attribute__((ext_vector_type(2))) float v2f;
typedef __attribute__((ext_vector_type(4))) float v4f;
typedef __attribute__((ext_vector_type(8))) float v8f;

#define D 128
#define BLOCK 256

#if defined(__gfx1250__) && __has_builtin(__builtin_amdgcn_global_load_async_to_lds_b128) && __has_builtin(__builtin_amdgcn_s_wait_asynccnt)
#define USE_ASYNC_LDS 1
typedef int v4i_t __attribute__((vector_size(4 * sizeof(int))));
typedef __attribute__((address_space(1))) v4i_t* gbl_v4i_ptr;
typedef __attribute__((address_space(3))) v4i_t* lds_v4i_ptr;
#else
#define USE_ASYNC_LDS 0
#endif

// ------------------------------------------------------------------ fill ----
__global__ void fill_kernel(float* __restrict__ p, float v, long n) {
  long i = (long)blockIdx.x * blockDim.x + threadIdx.x;
  if (i < n) p[i] = v;
}

// --------------------------------------------------------- gather helper ----
__device__ __forceinline__ void gather16B(const float* __restrict__ src,
                                          float* __restrict__ ldsDst) {
#if USE_ASYNC_LDS
  __builtin_amdgcn_global_load_async_to_lds_b128((gbl_v4i_ptr)src, (lds_v4i_ptr)ldsDst, 0, 0);
#else
  *(v4f*)ldsDst = *(const v4f*)src;
#endif
}

__device__ __forceinline__ void gather_fence() {
#if USE_ASYNC_LDS
  __builtin_amdgcn_s_wait_asynccnt(0);
#endif
  __syncthreads();
}

// ------------------------------------------------------- edge-style GEMM ----
// A row(e) = [ x[col[e]] | x[row[e]] | e_attr[e] | u[batch[row|col]] ]  (K = 512)
// SCATTER==0: out[e*D+n] = act(acc+bias)       (edge model)
// SCATTER==1: atomicAdd(&out[col[e]*D+n], ...) (message model -> agg)
template <int SCATTER>
__global__ __launch_bounds__(BLOCK)
void edge_gemm_kernel(const float* __restrict__ x,
                      const float* __restrict__ e_attr,
                      const float* __restrict__ u,
                      const int*   __restrict__ rowIdx,
                      const int*   __restrict__ colIdx,
                      const int*   __restrict__ batch,
                      const float* __restrict__ W,     // 512 x 128 row-major
                      const float* __restrict__ bias,  // 128
                      float*       __restrict__ out,
                      int E_, int use_col_batch, int act)
{
  const int K = 4 * D;        // 512
  const int STR = K + 4;      // 516 : stride%64==4 -> conflict-free WMMA A reads
  __shared__ float sA[16 * (4 * D + 4)];
  __shared__ int sRow[16], sCol[16], sG[16];

  const int tid = threadIdx.x;
  const int e0 = blockIdx.x * 16;

  // prefetch a hot slice of the (L2-resident) weight matrix
  __builtin_prefetch((const char*)W + (size_t)tid * 128, 0, 1);

  if (tid < 16) {
    int e = e0 + tid; if (e >= E_) e = E_ - 1;
    int r = rowIdx[e], c = colIdx[e];
    sRow[tid] = r; sCol[tid] = c;
    sG[tid] = batch[use_col_batch ? c : r];
  }
  __syncthreads();

  // gather-concat 16 x 512 tile into LDS, 16B per lane per step (async -> LDS)
  #pragma unroll
  for (int j = 0; j < (16 * (K / 4)) / BLOCK; ++j) {   // 8 steps
    int f  = j * BLOCK + tid;
    int el = f >> 7;                 // / (K/4)
    int k  = (f & 127) << 2;         // float offset within row
    const float* src;
    if (k < D)            src = x + (size_t)sCol[el] * D + k;
    else if (k < 2 * D)   src = x + (size_t)sRow[el] * D + (k - D);
    else if (k < 3 * D) {
      int e = e0 + el; if (e >= E_) e = E_ - 1;
      src = e_attr + (size_t)e * D + (k - 2 * D);
    } else                src = u + (size_t)sG[el] * D + (k - 3 * D);
    gather16B(src, &sA[el * STR + k]);
  }
  gather_fence();

  const int wave = tid >> 5, lane = tid & 31;
  const int ln = lane & 15, hi = lane >> 4;
  const int n = wave * 16 + ln;

  v8f acc;
  #pragma unroll
  for (int i = 0; i < 8; ++i) acc[i] = 0.0f;

  #pragma unroll 4
  for (int k0 = 0; k0 < K; k0 += 4) {
    int ka = k0 + 2 * hi;                  // lanes 0-15: K={k0,k0+1}; 16-31: {k0+2,k0+3}
    v2f a, b;
    a[0] = sA[ln * STR + ka];
    a[1] = sA[ln * STR + ka + 1];
    b[0] = W[(size_t)ka * D + n];
    b[1] = W[(size_t)(ka + 1) * D + n];
    acc = __builtin_amdgcn_wmma_f32_16x16x4_f32(false, a, false, b,
                                                (short)0, acc, false, false);
  }

  const float bv = bias[n];
  if (e0 + 16 <= E_) {                     // full tile: branchless epilogue
    float* po = out + (size_t)e0 * D + n;
    #pragma unroll
    for (int v = 0; v < 8; ++v) {
      int m = v + 8 * hi;                  // C layout: vgpr v -> row v (+8 hi half)
      float val = acc[v] + bv;
      if (act) val = fmaxf(val, 0.0f);
      if (SCATTER) atomicAdd(&out[(size_t)sCol[m] * D + n], val);
      else         po[(size_t)m * D] = val;
    }
  } else {                                 // tail tile (not hit for E%16==0)
    #pragma unroll
    for (int v = 0; v < 8; ++v) {
      int m = v + 8 * hi;
      int e = e0 + m;
      float val = acc[v] + bv;
      if (act) val = fmaxf(val, 0.0f);
      if (e < E_) {
        if (SCATTER) atomicAdd(&out[(size_t)sCol[m] * D + n], val);
        else         out[(size_t)e * D + n] = val;
      }
    }
  }
}

// --------------------------------------------------- 3-source GEMM (K=384) --
// A row(i) = [ s0[i] | s1[i] | u[g(i)] ],  g = batch ? batch[i] : i
__global__ __launch_bounds__(BLOCK)
void gemm3_kernel(const float* __restrict__ s0,
                  const float* __restrict__ s1,
                  const float* __restrict__ u,
                  const int*   __restrict__ batch_or_null,
                  const float* __restrict__ W,     // 384 x 128
                  const float* __restrict__ bias,
                  float*       __restrict__ out,
                  int M, int act)
{
  const int K = 3 * D;        // 384
  const int STR = K + 4;      // 388 : %64==4
  __shared__ float sA[16 * (3 * D + 4)];
  __shared__ int sG[16];

  const int tid = threadIdx.x;
  const int r0 = blockIdx.x * 16;

  __builtin_prefetch((const char*)W + (size_t)tid * 128, 0, 1);

  if (tid < 16) {
    int r = r0 + tid; if (r >= M) r = M - 1;
    sG[tid] = batch_or_null ? batch_or_null[r] : r;
  }
  __syncthreads();

  #pragma unroll
  for (int j = 0; j < (16 * (K / 4)) / BLOCK; ++j) {   // 6 steps
    int f  = j * BLOCK + tid;
    int rl = f / (K / 4);
    int k  = (f - rl * (K / 4)) << 2;
    int r  = r0 + rl; if (r >= M) r = M - 1;
    const float* src;
    if (k < D)          src = s0 + (size_t)r * D + k;
    else if (k < 2 * D) src = s1 + (size_t)r * D + (k - D);
    else                src = u + (size_t)sG[rl] * D + (k - 2 * D);
    gather16B(src, &sA[rl * STR + k]);
  }
  gather_fence();

  const int wave = tid >> 5, lane = tid & 31;
  const int ln = lane & 15, hi = lane >> 4;
  const int n = wave * 16 + ln;

  v8f acc;
  #pragma unroll
  for (int i = 0; i < 8; ++i) acc[i] = 0.0f;

  #pragma unroll 4
  for (int k0 = 0; k0 < K; k0 += 4) {
    int ka = k0 + 2 * hi;
    v2f a, b;
    a[0] = sA[ln * STR + ka];
    a[1] = sA[ln * STR + ka + 1];
    b[0] = W[(size_t)ka * D + n];
    b[1] = W[(size_t)(ka + 1) * D + n];
    acc = __builtin_amdgcn_wmma_f32_16x16x4_f32(false, a, false, b,
                                                (short)0, acc, false, false);
  }

  const float bv = bias[n];
  if (r0 + 16 <= M) {
    float* po = out + (size_t)r0 * D + n;
    #pragma unroll
    for (int v = 0; v < 8; ++v) {
      int m = v + 8 * hi;
      float val = acc[v] + bv;
      if (act) val = fmaxf(val, 0.0f);
      po[(size_t)m * D] = val;
    }
  } else {
    #pragma unroll
    for (int v = 0; v < 8; ++v) {
      int m = v + 8 * hi;
      int r = r0 + m;
      float val = acc[v] + bv;
      if (act) val = fmaxf(val, 0.0f);
      if (r < M) out[(size_t)r * D + n] = val;
    }
  }
}

// ------------------------------------------------------------- scatters -----
__global__ void node_scatter_kernel(const float* __restrict__ src,
                                    const int* __restrict__ batch,
                                    float* __restrict__ dst, int M) {
  long i = (long)blockIdx.x * blockDim.x + threadIdx.x;
  if (i >= (long)M * D) return;
  int r = (int)(i >> 7), c = (int)(i & (D - 1));
  atomicAdd(&dst[(size_t)batch[r] * D + c], src[i]);
}

__global__ void edge_scatter_kernel(const float* __restrict__ src,
                                    const int* __restrict__ colIdx,
                                    const int* __restrict__ batch,
                                    float* __restrict__ dst, int M) {
  long i = (long)blockIdx.x * blockDim.x + threadIdx.x;
  if (i >= (long)M * D) return;
  int r = (int)(i >> 7), c = (int)(i & (D - 1));
  atomicAdd(&dst[(size_t)batch[colIdx[r]] * D + c], src[i]);
}

// ------------------------------------------------------------- batchnorm ----
__global__ void bn_stats_kernel(const float* __restrict__ y,
                                float* __restrict__ stats, int M) {
  __shared__ float sh[512];
  int c = threadIdx.x & (D - 1);
  int h = threadIdx.x >> 7;     // 0/1
  int r0 = blockIdx.x * 64;
  float s = 0.f, s2 = 0.f;
  for (int r = h; r < 64; r += 2) {
    int rr = r0 + r;
    if (rr < M) { float v = y[(size_t)rr * D + c]; s += v; s2 += v * v; }
  }
  sh[threadIdx.x] = s;
  sh[256 + threadIdx.x] = s2;
  __syncthreads();
  if (h == 0) {
    atomicAdd(&stats[c],     sh[c]       + sh[D + c]);
    atomicAdd(&stats[D + c], sh[256 + c] + sh[256 + D + c]);
  }
}

__global__ void bn_apply_kernel(float* __restrict__ y,
                                const float* __restrict__ stats,
                                const float* __restrict__ g,
                                const float* __restrict__ b, int M) {
  long i = (long)blockIdx.x * blockDim.x + threadIdx.x;
  if (i >= (long)M * D) return;
  int c = (int)(i & (D - 1));
  float inv_m = 1.0f / (float)M;
  float mu  = stats[c] * inv_m;
  float var = stats[D + c] * inv_m - mu * mu;
  y[i] = (y[i] - mu) * rsqrtf(var + 1e-5f) * g[c] + b[c];
}

// ---------------------------------------------------------------- driver ----
extern "C" void kernel_launch(void* const* d_in, const int* in_sizes, int n_in,
                              void* d_out, int out_size, void* d_ws, size_t ws_size,
                              hipStream_t stream) {
  const float* x_in = (const float*)d_in[0];
  const int*   eidx = (const int*)d_in[1];
  const float* e_in = (const float*)d_in[2];
  const float* u_in = (const float*)d_in[3];
  const int*   batch = (const int*)d_in[4];

  const int N = in_sizes[0] / D;
  const int E = in_sizes[1] / 2;
  const int G = in_sizes[3] / D;
  const int* rowI = eidx;        // edge_index[0] (src)
  const int* colI = eidx + E;    // edge_index[1] (dest)

  // params: 5 + l*8 + {eW,eb,n1W,n1b,n2W,n2b,gW,gb}; bn: 29 + l*6 + {xg,xb,eg,eb,ug,ub}
  auto P  = [&](int l, int j) { return (const float*)d_in[5 + l * 8 + j]; };
  auto Bn = [&](int l, int j) { return (const float*)d_in[29 + l * 6 + j]; };

  float* X  = (float*)d_out;                 // N x D (final x)
  float* Eo = X + (size_t)N * D;             // E x D (final edge_attr)
  float* U  = Eo + (size_t)E * D;            // G x D (final u)

  float* agg   = (float*)d_ws;               // N x D
  float* nagg  = agg + (size_t)N * D;        // G x D
  float* eagg  = nagg + (size_t)G * D;       // G x D
  float* stats = eagg + (size_t)G * D;       // 2 x D

  auto zero = [&](float* p, long n) {
    fill_kernel<<<(int)((n + 255) / 256), 256, 0, stream>>>(p, 0.0f, n);
  };
  auto batchnorm = [&](float* y, int M, const float* g, const float* b) {
    zero(stats, 2 * D);
    bn_stats_kernel<<<(M + 63) / 64, BLOCK, 0, stream>>>(y, stats, M);
    bn_apply_kernel<<<(int)(((long)M * D + 255) / 256), 256, 0, stream>>>(y, stats, g, b, M);
  };

  const int egrid = (E + 15) / 16;
  const int ngrid = (N + 15) / 16;
  const int ggrid = (G + 15) / 16;

  for (int l = 0; l < 3; ++l) {
    const int act = (l != 2);
    const float* xin = (l == 0) ? x_in : X;
    const float* ein = (l == 0) ? e_in : Eo;
    const float* uin = (l == 0) ? u_in : U;

    // 1) edge model: e' = act([x[col]|x[row]|e|u[batch[row]]] @ eW + eb)
    edge_gemm_kernel<0><<<egrid, BLOCK, 0, stream>>>(
        xin, ein, uin, rowI, colI, batch, P(l, 0), P(l, 1), Eo, E, /*col_batch=*/0, act);

    // 2) message model + scatter-sum into agg (keyed by col)
    zero(agg, (long)N * D);
    edge_gemm_kernel<1><<<egrid, BLOCK, 0, stream>>>(
        xin, Eo, uin, rowI, colI, batch, P(l, 2), P(l, 3), agg, E, /*col_batch=*/1, act);

    // 3) node update: x' = act([x|agg|u[batch]] @ n2W + n2b)
    gemm3_kernel<<<ngrid, BLOCK, 0, stream>>>(
        xin, agg, uin, batch, P(l, 4), P(l, 5), X, N, act);

    // 4) graph aggregates
    zero(nagg, 2L * G * D);  // nagg + eagg contiguous
    node_scatter_kernel<<<(int)(((long)N * D + 255) / 256), 256, 0, stream>>>(X, batch, nagg, N);
    edge_scatter_kernel<<<(int)(((long)E * D + 255) / 256), 256, 0, stream>>>(Eo, colI, batch, eagg, E);

    // 5) global model: u' = act([nagg|eagg|u] @ gW + gb)
    gemm3_kernel<<<ggrid, BLOCK, 0, stream>>>(
        nagg, eagg, uin, (const int*)nullptr, P(l, 6), P(l, 7), U, G, act);

    // 6) batchnorm after non-final layers
    if (l != 2) {
      batchnorm(X,  N, Bn(l, 0), Bn(l, 1));
      batchnorm(Eo, E, Bn(l, 2), Bn(l, 3));
      batchnorm(U,  G, Bn(l, 4), Bn(l, 5));
    }
  }
}